// GCNWithPooling_30949534335548
// MI455X (gfx1250) — compile-verified
//
#include <hip/hip_runtime.h>
#include <hip/hip_bf16.h>

typedef __attribute__((ext_vector_type(2))) float v2f;
typedef __attribute__((ext_vector_type(8))) float v8f;

#define N_NODES 50000
#define N_EDGES 800000
#define HID     128
#define N_GRAPHS 256
#define BSTRIDE 136   // LDS row stride (floats): 2*136 % 64 = 16 -> half-waves hit disjoint banks

// ---------------------------------------------------------------------------
// Generic fill
// ---------------------------------------------------------------------------
__global__ void fill_kernel(float* __restrict__ p, float val, int n) {
    int i = blockIdx.x * blockDim.x + threadIdx.x;
    if (i < n) p[i] = val;
}

// ---------------------------------------------------------------------------
// Degree: deg[dst] += 1 for each real edge (deg pre-initialized to 1.0 for
// the self-loop contribution).
// ---------------------------------------------------------------------------
__global__ void degree_kernel(const int* __restrict__ ei, float* __restrict__ deg,
                              int n_edges) {
    int i = blockIdx.x * blockDim.x + threadIdx.x;
    if (i < n_edges) atomicAdd(&deg[ei[n_edges + i]], 1.0f);
}

__global__ void dinv_kernel(const float* __restrict__ deg, float* __restrict__ dinv,
                            int n) {
    int i = blockIdx.x * blockDim.x + threadIdx.x;
    if (i < n) {
        float d = deg[i];
        dinv[i] = d > 0.0f ? rsqrtf(d) : 0.0f;
    }
}

// ---------------------------------------------------------------------------
// FP32 WMMA GEMM: C[M,128] = A[M,128] @ B[128,128] (+bias, optional relu).
//
// Block = 256 threads = 8 waves. B[128,128] is staged once into LDS
// (padded stride 136 to kill half-wave bank conflicts), then each wave
// computes TPW row-tiles x one 16-col tile: one B fragment per K-step feeds
// TPW V_WMMA_F32_16X16X4_F32 ops (register-level B reuse).
// Requires M % (16*TPW) == 0; EXEC stays all-ones (WMMA requirement).
// ---------------------------------------------------------------------------
template<int TPW>
__launch_bounds__(256)
__global__ void gemm_wmma_f32(const float* __restrict__ A,
                              const float* __restrict__ B,
                              float* __restrict__ C,
                              const float* __restrict__ bias, int relu) {
    __shared__ float Bs[HID * BSTRIDE];

    const int tid = threadIdx.x;

    // Cooperative stage of B into LDS: 4096 float4 loads across 256 threads.
    for (int i = tid; i < HID * (HID / 4); i += 256) {
        const int row = i >> 5;          // /32 float4 per row
        const int c4  = (i & 31) << 2;
        const float4 v = *(const float4*)(B + row * HID + c4);
        *(float4*)(&Bs[row * BSTRIDE + c4]) = v;
    }
    __syncthreads();

    const int wave   = tid >> 5;
    const int lane   = tid & 31;
    const int m_base = blockIdx.x * (16 * TPW);
    const int n0     = wave * 16;
    const int r      = lane & 15;        // row (A) / col (B,C)
    const int kh     = lane >> 4;        // K half: 0 or 1

    v8f c[TPW];
    #pragma unroll
    for (int t = 0; t < TPW; ++t) c[t] = (v8f){};

    const float* aptr = A + (size_t)(m_base + r) * HID + 2 * kh;
    const float* bs   = Bs + (2 * kh) * BSTRIDE + (n0 + r);

    for (int k = 0; k < HID; k += 4) {
        v2f b;
        b.x = bs[0];
        b.y = bs[BSTRIDE];
        bs += 4 * BSTRIDE;
        #pragma unroll
        for (int t = 0; t < TPW; ++t) {
            const v2f a = *(const v2f*)(aptr + (size_t)t * 16 * HID);
            c[t] = __builtin_amdgcn_wmma_f32_16x16x4_f32(false, a, false, b,
                                                         (short)0, c[t], false, false);
        }
        aptr += 4;
    }

    const int col = n0 + r;
    const float bv = bias ? bias[col] : 0.0f;
    #pragma unroll
    for (int t = 0; t < TPW; ++t) {
        #pragma unroll
        for (int v = 0; v < 8; ++v) {
            const int row = m_base + t * 16 + v + 8 * kh;
            float val = c[t][v] + bv;
            if (relu) val = fmaxf(val, 0.0f);
            C[(size_t)row * HID + col] = val;
        }
    }
}

// ---------------------------------------------------------------------------
// Edge scatter: agg[dst] += h[src] * dinv[src]*dinv[dst].
// One wave32 per (edge or self-loop); lane handles 4 channels (float4 load,
// 4x global_atomic_add_f32). Working set is L2-resident (192 MB L2).
// ---------------------------------------------------------------------------
__global__ void scatter_kernel(const float* __restrict__ h,
                               const int* __restrict__ ei,
                               const float* __restrict__ dinv,
                               float* __restrict__ agg) {
    const int wid  = (blockIdx.x * blockDim.x + threadIdx.x) >> 5;
    const int lane = threadIdx.x & 31;
    const int total = N_EDGES + N_NODES;
    if (wid >= total) return;

    int s, d;
    if (wid < N_EDGES) {
        s = ei[wid];
        d = ei[N_EDGES + wid];
    } else {
        s = d = wid - N_EDGES;   // self loop
    }
    const float coeff = dinv[s] * dinv[d];

    const float4 v = ((const float4*)(h + (size_t)s * HID))[lane];
    float* ap = agg + (size_t)d * HID + lane * 4;
    atomicAdd(ap + 0, v.x * coeff);
    atomicAdd(ap + 1, v.y * coeff);
    atomicAdd(ap + 2, v.z * coeff);
    atomicAdd(ap + 3, v.w * coeff);
}

// ---------------------------------------------------------------------------
// In-place bias + ReLU over [M,128]
// ---------------------------------------------------------------------------
__global__ void bias_relu_kernel(float* __restrict__ h, const float* __restrict__ b,
                                 int n) {
    int i = blockIdx.x * blockDim.x + threadIdx.x;
    if (i < n) h[i] = fmaxf(h[i] + b[i & (HID - 1)], 0.0f);
}

// ---------------------------------------------------------------------------
// Global mean pool (accumulate): sums[batch[node]] += h[node]; cnt += 1.
// One wave per node.
// ---------------------------------------------------------------------------
__global__ void pool_kernel(const float* __restrict__ h, const int* __restrict__ batch,
                            float* __restrict__ sums, float* __restrict__ cnt) {
    const int node = (blockIdx.x * blockDim.x + threadIdx.x) >> 5;
    const int lane = threadIdx.x & 31;
    if (node >= N_NODES) return;
    const int g = batch[node];
    const float4 v = ((const float4*)(h + (size_t)node * HID))[lane];
    float* sp = sums + (size_t)g * HID + lane * 4;
    atomicAdd(sp + 0, v.x);
    atomicAdd(sp + 1, v.y);
    atomicAdd(sp + 2, v.z);
    atomicAdd(sp + 3, v.w);
    if (lane == 0) atomicAdd(&cnt[g], 1.0f);
}

__global__ void pool_finalize_kernel(const float* __restrict__ sums,
                                     const float* __restrict__ cnt,
                                     float* __restrict__ g) {
    int i = blockIdx.x * blockDim.x + threadIdx.x;
    if (i < N_GRAPHS * HID) {
        int gg = i >> 7;
        g[i] = sums[i] / fmaxf(cnt[gg], 1.0f);
    }
}

// ---------------------------------------------------------------------------
// Final head: out[g] = dot(g2[g,:], Wl2[:,0]) + bl2
// ---------------------------------------------------------------------------
__global__ void final_kernel(const float* __restrict__ g2, const float* __restrict__ Wl2,
                             const float* __restrict__ bl2, float* __restrict__ out) {
    int g = blockIdx.x * blockDim.x + threadIdx.x;
    if (g >= N_GRAPHS) return;
    float acc = bl2[0];
    #pragma unroll 8
    for (int c = 0; c < HID; ++c) acc += g2[(size_t)g * HID + c] * Wl2[c];
    out[g] = acc;
}

// ---------------------------------------------------------------------------
extern "C" void kernel_launch(void* const* d_in, const int* in_sizes, int n_in,
                              void* d_out, int out_size, void* d_ws, size_t ws_size,
                              hipStream_t stream) {
    const float* x   = (const float*)d_in[0];
    const int*   ei  = (const int*)d_in[1];
    const int*   bat = (const int*)d_in[2];
    const float* W1  = (const float*)d_in[3];
    const float* b1  = (const float*)d_in[4];
    const float* W2  = (const float*)d_in[5];
    const float* b2  = (const float*)d_in[6];
    const float* Wl1 = (const float*)d_in[7];
    const float* bl1 = (const float*)d_in[8];
    const float* Wl2 = (const float*)d_in[9];
    const float* bl2 = (const float*)d_in[10];
    float* out = (float*)d_out;

    // Workspace layout (floats)
    float* buf0 = (float*)d_ws;                        // GEMM outputs  [N,128]
    float* buf1 = buf0 + (size_t)N_NODES * HID;        // scatter accum [N,128]
    float* deg  = buf1 + (size_t)N_NODES * HID;        // [N]
    float* dinv = deg + N_NODES;                       // [N]
    float* sums = dinv + N_NODES;                      // [G,128]
    float* cnt  = sums + (size_t)N_GRAPHS * HID;       // [G]
    float* gbuf = cnt + N_GRAPHS;                      // [G,128]
    float* g2   = gbuf + (size_t)N_GRAPHS * HID;       // [G,128]

    const int NB = 256;
    const int nh = N_NODES * HID;
    const int big_blocks = N_NODES / 80;               // 625 blocks x 80 rows (TPW=5)

    // --- degree / normalization ---
    fill_kernel<<<(N_NODES + NB - 1) / NB, NB, 0, stream>>>(deg, 1.0f, N_NODES); // self-loops
    degree_kernel<<<(N_EDGES + NB - 1) / NB, NB, 0, stream>>>(ei, deg, N_EDGES);
    dinv_kernel<<<(N_NODES + NB - 1) / NB, NB, 0, stream>>>(deg, dinv, N_NODES);

    // --- layer 1 ---
    fill_kernel<<<(nh + NB - 1) / NB, NB, 0, stream>>>(buf1, 0.0f, nh);
    gemm_wmma_f32<5><<<big_blocks, NB, 0, stream>>>(x, W1, buf0, nullptr, 0);
    scatter_kernel<<<(N_EDGES + N_NODES) / 8, NB, 0, stream>>>(buf0, ei, dinv, buf1);
    bias_relu_kernel<<<(nh + NB - 1) / NB, NB, 0, stream>>>(buf1, b1, nh);

    // --- layer 2 (re-zero buf1 after GEMM consumed it; stream order ensures safety) ---
    gemm_wmma_f32<5><<<big_blocks, NB, 0, stream>>>(buf1, W2, buf0, nullptr, 0);
    fill_kernel<<<(nh + NB - 1) / NB, NB, 0, stream>>>(buf1, 0.0f, nh);
    scatter_kernel<<<(N_EDGES + N_NODES) / 8, NB, 0, stream>>>(buf0, ei, dinv, buf1);
    bias_relu_kernel<<<(nh + NB - 1) / NB, NB, 0, stream>>>(buf1, b2, nh);

    // --- pooling ---
    fill_kernel<<<(N_GRAPHS * HID + NB - 1) / NB, NB, 0, stream>>>(sums, 0.0f, N_GRAPHS * HID);
    fill_kernel<<<1, NB, 0, stream>>>(cnt, 0.0f, N_GRAPHS);
    pool_kernel<<<(N_NODES * 32 + NB - 1) / NB, NB, 0, stream>>>(buf1, bat, sums, cnt);
    pool_finalize_kernel<<<(N_GRAPHS * HID + NB - 1) / NB, NB, 0, stream>>>(sums, cnt, gbuf);

    // --- head (M=256 = 16 tiles, TPW=1) ---
    gemm_wmma_f32<1><<<N_GRAPHS / 16, NB, 0, stream>>>(gbuf, Wl1, g2, bl1, 1);
    final_kernel<<<1, NB, 0, stream>>>(g2, Wl2, bl2, out);
}